// LogSimpleSlater_5377299054807
// MI455X (gfx1250) — compile-verified
//
#include <hip/hip_runtime.h>

// One workgroup (256 threads = 8 wave32) per walker.
// Slater matrix built and LU-factored entirely in LDS (66 KB of the 320 KB/WGP).
// L21 is stored NEGATED so the Schur update D = (-L)*U + C needs no VALU
// negation on the WMMA A-fragment (f32 WMMA has no A-neg modifier).
// Schur complement updates use V_WMMA_F32_16X16X4_F32 (full-f32 matrix pipe).

#define NMAT   128
#define LDM    129            // padded LDS row stride (floats) -> conflict-free columns
#define PANEL  16
#define NPAN   (NMAT / PANEL) // 8
#define TPB    256

typedef float v2f __attribute__((ext_vector_type(2)));
typedef float v8f __attribute__((ext_vector_type(8)));

__launch_bounds__(TPB)
__global__ void slater_slogdet_kernel(const float* __restrict__ rs,
                                      const float* __restrict__ kpoints,
                                      float* __restrict__ out)
{
    extern __shared__ float smem[];
    float* A  = smem;                 // NMAT * LDM
    float* kx = A + NMAT * LDM;       // 128
    float* ky = kx + NMAT;            // 128
    float* kz = ky + NMAT;            // 128
    float* rb = kz + NMAT;            // 3 * 128
    __shared__ int   s_piv;
    __shared__ float s_pval;

    const int tid  = threadIdx.x;
    const int b    = blockIdx.x;
    const int lane = tid & 31;
    const int wave = tid >> 5;

    // ---- stage kpoints (transposed) and this walker's electron coords ----
    if (tid < NMAT) {
        kx[tid] = kpoints[tid * 3 + 0];
        ky[tid] = kpoints[tid * 3 + 1];
        kz[tid] = kpoints[tid * 3 + 2];
        const float* rp = rs + ((size_t)b * NMAT + tid) * 3;
        rb[tid * 3 + 0] = rp[0];
        rb[tid * 3 + 1] = rp[1];
        rb[tid * 3 + 2] = rp[2];
    }
    __syncthreads();

    // ---- build Slater matrix in LDS ----
    // S[i][j] = cos(k_j . r_i) for j==0 or j odd; sin(k_j . r_i) for even j>=2
    for (int e = tid; e < NMAT * NMAT; e += TPB) {
        const int i = e >> 7;
        const int j = e & (NMAT - 1);
        const float dot = kx[j] * rb[i * 3 + 0]
                        + ky[j] * rb[i * 3 + 1]
                        + kz[j] * rb[i * 3 + 2];
        float s, c;
        __sincosf(dot, &s, &c);
        const bool use_cos = (j == 0) || (j & 1);
        A[i * LDM + j] = use_cos ? c : s;
    }
    __syncthreads();

    float logacc = 0.0f;  // every thread keeps the same value; thread 0 writes it

    for (int p = 0; p < NPAN; ++p) {
        const int c0 = p * PANEL;

        // ================= panel factorization (partial pivoting) =================
        for (int j = c0; j < c0 + PANEL; ++j) {
            // --- pivot search over column j, rows [j,127]: wave 0 only ---
            if (wave == 0) {
                float best = -1.0f;
                int   bi   = j;
                for (int i = j + lane; i < NMAT; i += 32) {
                    const float v = fabsf(A[i * LDM + j]);
                    if (v > best) { best = v; bi = i; }
                }
                #pragma unroll
                for (int off = 16; off > 0; off >>= 1) {
                    const float ob = __shfl_xor(best, off, 32);
                    const int   oi = __shfl_xor(bi,   off, 32);
                    if (ob > best) { best = ob; bi = oi; }
                }
                if (lane == 0) { s_piv = bi; s_pval = best; }
            }
            __syncthreads();

            const int piv = s_piv;
            if (piv != j && tid < NMAT) {           // full row swap (sign irrelevant)
                const float t0     = A[j   * LDM + tid];
                A[j   * LDM + tid] = A[piv * LDM + tid];
                A[piv * LDM + tid] = t0;
            }
            __syncthreads();

            logacc += __logf(s_pval);               // log|U_jj| = log(pivot magnitude)

            // --- scale column (store -L!) + rank-1 update restricted to the panel ---
            const float ninv = -1.0f / A[j * LDM + j];
            const int i = j + 1 + tid;              // <=127 rows -> one row per thread
            if (i < NMAT) {
                float* Ai = &A[i * LDM];
                const float* Uj = &A[j * LDM];
                const float nlij = Ai[j] * ninv;    // -l_ij
                Ai[j] = nlij;
                for (int l = j + 1; l < c0 + PANEL; ++l)
                    Ai[l] = fmaf(nlij, Uj[l], Ai[l]);   // A -= l*U  (broadcast read)
            }
            __syncthreads();
        }

        const int t0col = c0 + PANEL;
        const int R = NMAT - t0col;                 // trailing size
        if (R > 0) {
            // ============ triangular solve: U12 = L11^{-1} * A12 ============
            // L11 strictly-lower entries are stored negated -> forward-sub uses +=
            if (tid < R) {
                const int t = t0col + tid;
                float y[PANEL];
                #pragma unroll
                for (int m = 0; m < PANEL; ++m) y[m] = A[(c0 + m) * LDM + t];
                #pragma unroll
                for (int m = 0; m < PANEL; ++m) {
                    const float ym = y[m];
                    for (int jj = m + 1; jj < PANEL; ++jj)
                        y[jj] = fmaf(A[(c0 + jj) * LDM + (c0 + m)], ym, y[jj]);
                }
                #pragma unroll
                for (int m = 0; m < PANEL; ++m) A[(c0 + m) * LDM + t] = y[m];
            }
            __syncthreads();

            // ===== Schur update: A22 += (-L21) * U12 via WMMA f32 16x16x4 =====
            const int nt   = R >> 4;                // 16x16 tiles per dimension
            const int half = lane >> 4;             // 0: lanes 0-15, 1: lanes 16-31
            const int lm   = lane & 15;
            for (int tile = wave; tile < nt * nt; tile += (TPB / 32)) { // wave-uniform
                const int tm  = tile / nt;
                const int tn  = tile % nt;
                const int rb0 = t0col + tm * 16;
                const int cb0 = t0col + tn * 16;

                // C/D layout: VGPR v -> row M = v + 8*half, col N = lm
                v8f c;
                #pragma unroll
                for (int v = 0; v < 8; ++v)
                    c[v] = A[(rb0 + v + 8 * half) * LDM + cb0 + lm];

                // K = 16 as four 16x16x4 steps; -L already stored in LDS
                #pragma unroll
                for (int kk = 0; kk < 4; ++kk) {
                    const int k0 = c0 + kk * 4 + 2 * half;  // K pair for this lane-half
                    v2f a, bf;
                    a[0]  = A[(rb0 + lm) * LDM + k0];       // A: lane=M, vgpr/half=K
                    a[1]  = A[(rb0 + lm) * LDM + k0 + 1];
                    bf[0] = A[(k0    ) * LDM + cb0 + lm];   // B: lane=N, vgpr/half=K
                    bf[1] = A[(k0 + 1) * LDM + cb0 + lm];
                    c = __builtin_amdgcn_wmma_f32_16x16x4_f32(
                            false, a, false, bf, (short)0, c, false, false);
                }

                #pragma unroll
                for (int v = 0; v < 8; ++v)
                    A[(rb0 + v + 8 * half) * LDM + cb0 + lm] = c[v];
            }
            __syncthreads();
        }
    }

    if (tid == 0) out[b] = logacc;
}

extern "C" void kernel_launch(void* const* d_in, const int* in_sizes, int n_in,
                              void* d_out, int out_size, void* d_ws, size_t ws_size,
                              hipStream_t stream) {
    (void)n_in; (void)d_ws; (void)ws_size; (void)out_size;
    const float* rs = (const float*)d_in[0];   // [B, 128, 3] f32
    const float* kp = (const float*)d_in[1];   // [128, 3]    f32
    float* out = (float*)d_out;                // [B]         f32
    const int B = in_sizes[0] / (NMAT * 3);
    const size_t shmem = (size_t)(NMAT * LDM + 3 * NMAT + 3 * NMAT) * sizeof(float);
    slater_slogdet_kernel<<<dim3(B), dim3(TPB), shmem, stream>>>(rs, kp, out);
}